// MambaBlock_5686536700303
// MI455X (gfx1250) — compile-verified
//
#include <hip/hip_runtime.h>
#include <hip/hip_bf16.h>
#include <math.h>

// ---- model dims ----
#define BATCH     2
#define LSEQ      2048
#define DMODEL    768
#define DSTATE    16
#define DCONV     4
#define DINNER    1536            // 2*768
#define DTRANK    48              // (768+15)/16
#define NROWS     (BATCH*LSEQ)    // 4096
#define XZCOLS    (2*DINNER)      // 3072
#define PROJCOLS  (DTRANK + 2*DSTATE) // 80

typedef __attribute__((ext_vector_type(2))) float v2f;
typedef __attribute__((ext_vector_type(8))) float v8f;

// =====================================================================
// LayerNorm: one row (768) per block of 256 threads
// =====================================================================
__global__ void ln_kernel(const float* __restrict__ x,
                          const float* __restrict__ gamma,
                          const float* __restrict__ beta,
                          float* __restrict__ h) {
  __shared__ float red[256];
  const int row = blockIdx.x;
  const int tid = threadIdx.x;
  const float* xr = x + (size_t)row * DMODEL;
  float v0 = xr[tid], v1 = xr[tid + 256], v2 = xr[tid + 512];

  red[tid] = v0 + v1 + v2;
  __syncthreads();
  for (int s = 128; s > 0; s >>= 1) {
    if (tid < s) red[tid] += red[tid + s];
    __syncthreads();
  }
  const float mu = red[0] * (1.0f / DMODEL);
  __syncthreads();

  float c0 = v0 - mu, c1 = v1 - mu, c2 = v2 - mu;
  red[tid] = c0 * c0 + c1 * c1 + c2 * c2;
  __syncthreads();
  for (int s = 128; s > 0; s >>= 1) {
    if (tid < s) red[tid] += red[tid + s];
    __syncthreads();
  }
  const float rstd = rsqrtf(red[0] * (1.0f / DMODEL) + 1e-5f);

  float* hr = h + (size_t)row * DMODEL;
  hr[tid]       = c0 * rstd * gamma[tid]       + beta[tid];
  hr[tid + 256] = c1 * rstd * gamma[tid + 256] + beta[tid + 256];
  hr[tid + 512] = c2 * rstd * gamma[tid + 512] + beta[tid + 512];
}

// =====================================================================
// f32 WMMA GEMM:  C[M,N] = A[M,K] @ B[K,N]  (+ epilogue)
//   mode 0: plain   mode 1: softplus(C + bias[N])   mode 2: C + resid
// Block = 128 threads (4 waves). Each block: 16(M) x 64(N) tile.
// Wave w handles N-subtile w. K chunked by 64 through LDS.
// B-tile staging uses GLOBAL_LOAD_ASYNC_TO_LDS_B32 (ASYNCcnt path) with
// per-lane LDS destinations, which performs the row->col transpose
// scatter directly; falls back to guarded scalar staging on edge tiles.
// WMMA layout (V_WMMA_F32_16X16X4_F32):
//   A: lanes 0-15 m=lane, v={K0,K1}; lanes 16-31 m=lane-16, v={K2,K3}
//   B: lanes 0-15 n=lane, v={K0,K1}; lanes 16-31 n=lane-16, v={K2,K3}
//   C: vgpr r: lanes 0-15 (m=r, n=lane); lanes 16-31 (m=r+8, n=lane-16)
// =====================================================================
#define KC 64
__global__ void gemm_wmma_f32(const float* __restrict__ A,
                              const float* __restrict__ B,
                              float* __restrict__ C,
                              int M, int N, int K,
                              int lda, int ldb, int ldc,
                              int mode, const float* __restrict__ aux) {
  __shared__ float As[16][KC + 2];
  __shared__ float BsT[64][KC + 2];

  const int tid  = threadIdx.x;
  const int lane = tid & 31;
  const int wave = tid >> 5;          // 0..3
  const int m0   = blockIdx.x * 16;
  const int n0   = blockIdx.y * 64;
  const int mrow = lane & 15;
  const int koff = (lane >> 4) * 2;   // 0 or 2

  v8f acc = {};

  for (int kb = 0; kb < K; kb += KC) {
    // ---- stage A chunk (16 x KC), small: plain guarded loads ----
    for (int i = tid; i < 16 * KC; i += 128) {
      int r = i >> 6, c = i & 63;
      int gk = kb + c;
      As[r][c] = (gk < K) ? A[(size_t)(m0 + r) * lda + gk] : 0.0f;
    }

    // ---- stage B chunk transposed: BsT[n_local][k_local] ----
    const bool full = (kb + KC <= K) && (n0 + 64 <= N);
    if (full) {
      // async global->LDS copy; each lane scatters one dword to its own
      // LDS address (transpose happens in the address mapping)
      for (int i = tid; i < 64 * KC; i += 128) {
        int kk = i >> 6, c = i & 63;
        const float* gp = &B[(size_t)(kb + kk) * ldb + (n0 + c)];
        unsigned lp = (unsigned)(reinterpret_cast<uintptr_t>(&BsT[c][kk]));
        asm volatile("global_load_async_to_lds_b32 %0, %1, off"
                     :: "v"(lp), "v"(gp)
                     : "memory");
      }
      asm volatile("s_wait_asynccnt 0" ::: "memory");
    } else {
      for (int i = tid; i < 64 * KC; i += 128) {
        int kk = i >> 6, c = i & 63;
        int gk = kb + kk, gc = n0 + c;
        BsT[c][kk] = (gk < K && gc < N) ? B[(size_t)gk * ldb + gc] : 0.0f;
      }
    }

    // prefetch next K-chunk of A (emits global_prefetch_b8)
    if (kb + KC < K) {
      __builtin_prefetch(&A[(size_t)(m0 + (tid & 15)) * lda + kb + KC], 0, 0);
    }
    __syncthreads();

#pragma unroll
    for (int k0 = 0; k0 < KC; k0 += 4) {
      v2f a  = *(const v2f*)&As[mrow][k0 + koff];
      v2f bb = *(const v2f*)&BsT[(wave << 4) + mrow][k0 + koff];
      acc = __builtin_amdgcn_wmma_f32_16x16x4_f32(
          /*neg_a=*/false, a, /*neg_b=*/false, bb,
          /*c_mod=*/(short)0, acc, /*reuse_a=*/false, /*reuse_b=*/false);
    }
    __syncthreads();
  }

  const int col = n0 + (wave << 4) + (lane & 15);
  if (col < N) {
#pragma unroll
    for (int r = 0; r < 8; ++r) {
      int row = m0 + r + ((lane >> 4) << 3);
      float v = acc[r];
      if (mode == 1) {              // softplus(v + bias)
        v += aux[col];
        v = (v > 20.0f) ? v : log1pf(__expf(v));
      } else if (mode == 2) {       // residual add
        v += aux[(size_t)row * ldc + col];
      }
      C[(size_t)row * ldc + col] = v;
    }
  }
}

// =====================================================================
// Depthwise causal conv (width 4) + bias + SiLU.
// xin = xz[:, :DINNER] (row stride XZCOLS). Output xc[NROWS, DINNER].
// =====================================================================
__global__ void conv_silu_kernel(const float* __restrict__ xz,
                                 const float* __restrict__ conv_w,
                                 const float* __restrict__ conv_b,
                                 float* __restrict__ xc) {
  const size_t idx = (size_t)blockIdx.x * blockDim.x + threadIdx.x;
  if (idx >= (size_t)NROWS * DINNER) return;
  const int d   = (int)(idx % DINNER);
  const int row = (int)(idx / DINNER);
  const int l   = row % LSEQ;
  const int b   = row / LSEQ;

  float accv = conv_b[d];
#pragma unroll
  for (int k = 0; k < DCONV; ++k) {
    int lsrc = l - (DCONV - 1) + k;
    if (lsrc >= 0)
      accv += xz[((size_t)(b * LSEQ + lsrc)) * XZCOLS + d] * conv_w[d * DCONV + k];
  }
  accv = accv / (1.0f + __expf(-accv));   // SiLU
  xc[(size_t)row * DINNER + d] = accv;
}

// =====================================================================
// Selective scan. One lane per (channel d, state s). 16 channels/block.
// grid = (DINNER/16, BATCH), block = 256.
// Reads dt, xc (coalesced), B/C staged from proj into LDS in 128-step
// chunks. Reduces y over 16 state lanes with __shfl_xor.
// Epilogue (lane s==0): y += x*D;  y *= silu(z);  write IN PLACE to xc.
// =====================================================================
#define SCHUNK 128
__global__ void scan_kernel(const float* __restrict__ proj,
                            const float* __restrict__ dt,
                            float* __restrict__ xc,     // read x_t, write gated y
                            const float* __restrict__ xz,   // z = xz[:, DINNER:]
                            const float* __restrict__ A_log,
                            const float* __restrict__ D_param) {
  __shared__ float bc[SCHUNK][32];   // [step][ B(0..15) | C(16..31) ]

  const int tid = threadIdx.x;
  const int s   = tid & 15;
  const int chl = tid >> 4;                    // 0..15 local channel
  const int d   = blockIdx.x * 16 + chl;
  const int b   = blockIdx.y;

  const float Aneg = -__expf(A_log[d * DSTATE + s]);
  const float Dp   = D_param[d];

  float hstate = 0.0f;

  for (int c0 = 0; c0 < LSEQ; c0 += SCHUNK) {
    // cooperative stage of B/C for SCHUNK timesteps
    for (int i = tid; i < SCHUNK * 32; i += 256) {
      int ll = i >> 5, j = i & 31;
      bc[ll][j] = proj[((size_t)(b * LSEQ + c0 + ll)) * PROJCOLS + DTRANK + j];
    }
    __syncthreads();

    for (int ll = 0; ll < SCHUNK; ++ll) {
      const size_t row = (size_t)b * LSEQ + c0 + ll;
      const float dt_t = dt[row * DINNER + d];
      const float x_t  = xc[row * DINNER + d];
      const float dA   = __expf(dt_t * Aneg);
      hstate = hstate * dA + (dt_t * x_t) * bc[ll][s];
      float yp = hstate * bc[ll][16 + s];
      // reduce across the 16 state lanes
      yp += __shfl_xor(yp, 1, 16);
      yp += __shfl_xor(yp, 2, 16);
      yp += __shfl_xor(yp, 4, 16);
      yp += __shfl_xor(yp, 8, 16);
      if (s == 0) {
        float y  = yp + x_t * Dp;
        float zv = xz[row * XZCOLS + DINNER + d];
        y *= zv / (1.0f + __expf(-zv));      // * silu(z)
        xc[row * DINNER + d] = y;            // in-place gated output
      }
    }
    __syncthreads();
  }
}

// =====================================================================
// Host launcher
// =====================================================================
extern "C" void kernel_launch(void* const* d_in, const int* in_sizes, int n_in,
                              void* d_out, int out_size, void* d_ws, size_t ws_size,
                              hipStream_t stream) {
  const float* x      = (const float*)d_in[0];
  const float* gamma  = (const float*)d_in[1];
  const float* beta   = (const float*)d_in[2];
  const float* W_in   = (const float*)d_in[3];
  const float* conv_w = (const float*)d_in[4];
  const float* conv_b = (const float*)d_in[5];
  const float* W_x    = (const float*)d_in[6];
  const float* W_dt   = (const float*)d_in[7];
  const float* b_dt   = (const float*)d_in[8];
  const float* A_log  = (const float*)d_in[9];
  const float* D_par  = (const float*)d_in[10];
  const float* W_out  = (const float*)d_in[11];
  float* out = (float*)d_out;

  float* ws = (float*)d_ws;
  // workspace layout (floats)
  float* h_buf  = ws;                                    // 4096*768
  float* xz_buf = h_buf  + (size_t)NROWS * DMODEL;       // 4096*3072
  float* xc_buf = xz_buf + (size_t)NROWS * XZCOLS;       // 4096*1536 (also gated y)
  float* pr_buf = xc_buf + (size_t)NROWS * DINNER;       // 4096*80
  float* dt_buf = pr_buf + (size_t)NROWS * PROJCOLS;     // 4096*1536

  // 1) LayerNorm
  ln_kernel<<<NROWS, 256, 0, stream>>>(x, gamma, beta, h_buf);

  // 2) xz = h @ W_in   [4096,768]x[768,3072]
  gemm_wmma_f32<<<dim3(NROWS / 16, XZCOLS / 64), 128, 0, stream>>>(
      h_buf, W_in, xz_buf, NROWS, XZCOLS, DMODEL, DMODEL, XZCOLS, XZCOLS, 0, nullptr);

  // 3) depthwise conv + SiLU -> xc
  {
    size_t n = (size_t)NROWS * DINNER;
    conv_silu_kernel<<<(unsigned)((n + 255) / 256), 256, 0, stream>>>(
        xz_buf, conv_w, conv_b, xc_buf);
  }

  // 4) proj = xc @ W_x   [4096,1536]x[1536,80]
  gemm_wmma_f32<<<dim3(NROWS / 16, (PROJCOLS + 63) / 64), 128, 0, stream>>>(
      xc_buf, W_x, pr_buf, NROWS, PROJCOLS, DINNER, DINNER, PROJCOLS, PROJCOLS, 0, nullptr);

  // 5) dt = softplus(proj[:, :48] @ W_dt + b_dt)   [4096,48]x[48,1536]
  gemm_wmma_f32<<<dim3(NROWS / 16, DINNER / 64), 128, 0, stream>>>(
      pr_buf, W_dt, dt_buf, NROWS, DINNER, DTRANK, PROJCOLS, DINNER, DINNER, 1, b_dt);

  // 6) selective scan + D skip + z-gate (in place over xc)
  scan_kernel<<<dim3(DINNER / 16, BATCH), 256, 0, stream>>>(
      pr_buf, dt_buf, xc_buf, xz_buf, A_log, D_par);

  // 7) out = x + y_gated @ W_out   [4096,1536]x[1536,768]
  gemm_wmma_f32<<<dim3(NROWS / 16, DMODEL / 64), 128, 0, stream>>>(
      xc_buf, W_out, out, NROWS, DMODEL, DINNER, DINNER, DMODEL, DMODEL, 2, x);
}